// EventTempRel_poincare_14886356648169
// MI455X (gfx1250) — compile-verified
//
#include <hip/hip_runtime.h>
#include <hip/hip_bf16.h>
#include <math.h>

typedef float v2f __attribute__((ext_vector_type(2)));
typedef float v8f __attribute__((ext_vector_type(8)));

#define EPS_F 1e-15f
#define BND_F 0.99999994f   // largest float < 1.0; matches the 1-1e-7 clamp intent

__device__ __forceinline__ float artanh_clamped(float x) {
    x = fminf(fmaxf(x, -BND_F), BND_F);
    return 0.5f * logf((1.0f + x) / (1.0f - x));
}

// ---------------------------------------------------------------------------
// Kernel A: for each of the 384 selected rows, find the one-hot index,
// gather the H=768 vector, apply expmap0 (tanh(||x||)/||x|| * x), store to Y.
// One block (256 threads) per row; 3 elements per thread.
// ---------------------------------------------------------------------------
__global__ void gather_expmap_kernel(const float* __restrict__ encoded,
                                     const float* __restrict__ n_encoded,
                                     const float* __restrict__ mask1,
                                     const float* __restrict__ mask2,
                                     const float* __restrict__ maskn,
                                     float* __restrict__ Y,
                                     float* __restrict__ xn,
                                     int B, int S, int H) {
    const int r = blockIdx.x;      // 0..383
    const int t = threadIdx.x;     // 0..255 (== S)

    const float* mask;
    const float* src;
    if (r < B) {
        mask = mask1 + (size_t)r * S;
        src  = encoded + (size_t)r * S * H;
    } else if (r < 2 * B) {
        const int b = r - B;
        mask = mask2 + (size_t)b * S;
        src  = encoded + (size_t)b * S * H;
    } else {
        const int q = r - 2 * B;   // 0..255
        mask = maskn + (size_t)q * S;
        src  = n_encoded + (size_t)q * S * H;
    }

    __shared__ int   s_idx;
    __shared__ float red[256];
    if (t == 0) s_idx = 0;
    __syncthreads();
    if (mask[t] > 0.5f) s_idx = t;   // exactly one lane writes (one-hot)
    __syncthreads();

    const float* x = src + (size_t)s_idx * H;

    float xv[3];
    float local = 0.0f;
    #pragma unroll
    for (int i = 0; i < 3; ++i) {
        xv[i] = x[t + i * 256];
        local += xv[i] * xv[i];
    }
    red[t] = local;
    __syncthreads();
    for (int off = 128; off > 0; off >>= 1) {
        if (t < off) red[t] += red[t + off];
        __syncthreads();
    }
    float n  = fmaxf(sqrtf(red[0]), EPS_F);
    float sc = tanhf(n) / n;
    #pragma unroll
    for (int i = 0; i < 3; ++i)
        Y[(size_t)r * H + t + i * 256] = sc * xv[i];
    if (t == 0) xn[r] = fmaxf(tanhf(n), EPS_F);   // ||expmap0(x)|| == tanh(||x||)
}

// ---------------------------------------------------------------------------
// Kernel B: MX[384,64] = Y[384,768] @ W[64,768]^T via V_WMMA_F32_16X16X4_F32.
// One wave32 per 16x16 output tile; K-loop over 768 in steps of 4.
// A-layout (16x4 f32): lane m = lane&15, VGPR pair holds K = koff, koff+1
// with koff = 0 (lanes 0-15) or 2 (lanes 16-31).   B (4x16): symmetric, N=lane&15.
// C/D (16x16 f32): VGPR i holds row (i + 8*(lane>=16)), col = lane&15.
// ---------------------------------------------------------------------------
__global__ void wmma_gemm_kernel(const float* __restrict__ Y,   // [R, H]
                                 const float* __restrict__ W,   // [D, H]
                                 float* __restrict__ MX,        // [R, D]
                                 int H, int D) {
    const int tile = blockIdx.x;       // 0..95
    const int nt   = tile & 3;         // D/16 = 4 tiles along N
    const int mt   = tile >> 2;        // 24 tiles along M
    const int lane = threadIdx.x;      // 0..31, full wave => EXEC all 1s

    const int mrow = mt * 16 + (lane & 15);
    const int ncol = nt * 16 + (lane & 15);
    const int koff = (lane >> 4) << 1;  // 0 or 2

    const float* arow = Y + (size_t)mrow * H;
    const float* brow = W + (size_t)ncol * H;   // B[k][n] = W[n][k]

    v8f c = {0.f, 0.f, 0.f, 0.f, 0.f, 0.f, 0.f, 0.f};
    for (int k = 0; k < H; k += 4) {
        v2f a = { arow[k + koff], arow[k + koff + 1] };
        v2f b = { brow[k + koff], brow[k + koff + 1] };
        c = __builtin_amdgcn_wmma_f32_16x16x4_f32(
                /*neg_a=*/false, a, /*neg_b=*/false, b,
                /*c_mod=*/(short)0, c, /*reuse_a=*/false, /*reuse_b=*/false);
    }

    const int rbase = mt * 16 + ((lane >> 4) << 3);  // +8 for upper half-wave
    #pragma unroll
    for (int i = 0; i < 8; ++i)
        MX[(size_t)(rbase + i) * D + ncol] = c[i];
}

// ---------------------------------------------------------------------------
// Kernel C: mobius_matvec rescale, in place on MX.
// out = tanh(mn/xn * artanh(xn)) * mx / mn,   mn = ||mx||.
// One block (64 threads) per row.
// ---------------------------------------------------------------------------
__global__ void mobius_scale_kernel(float* __restrict__ MX,
                                    const float* __restrict__ xn,
                                    int D) {
    const int r = blockIdx.x;
    const int t = threadIdx.x;   // 0..63
    __shared__ float red[64];
    float v = MX[(size_t)r * D + t];
    red[t] = v * v;
    __syncthreads();
    for (int off = 32; off > 0; off >>= 1) {
        if (t < off) red[t] += red[t + off];
        __syncthreads();
    }
    float mn = fmaxf(sqrtf(red[0]), EPS_F);
    float x  = xn[r];
    float s  = tanhf((mn / x) * artanh_clamped(x)) / mn;
    MX[(size_t)r * D + t] = s * v;
}

// ---------------------------------------------------------------------------
// Kernel D: final loss. Closed form: for mobius_add(-x, y),
// num_i = A*(-x_i) + Bc*y_i with A = 1 - 2<x,y> + ||y||², Bc = 1 - ||x||²,
// so ||num||² = A²||x||² - 2 A Bc <x,y> + Bc²||y||² — only dot products needed.
// One block, 64 threads (one per batch), LDS mean reduction.
// ---------------------------------------------------------------------------
__device__ __forceinline__ float pdist_from_dots(float x2, float y2, float dxy) {
    float A   = 1.0f - 2.0f * dxy + y2;
    float Bc  = 1.0f - x2;
    float den = fmaxf(1.0f - 2.0f * dxy + x2 * y2, EPS_F);
    float n2  = fmaxf(A * A * x2 - 2.0f * A * Bc * dxy + Bc * Bc * y2, 0.0f);
    return 2.0f * artanh_clamped(sqrtf(n2) / den);
}

__global__ void final_loss_kernel(const float* __restrict__ U,  // [384, D]
                                  float* __restrict__ out,
                                  int B, int D, int NEG) {
    const int b = threadIdx.x;   // 0..63
    const float* u = U + (size_t)b * D;
    const float* v = U + (size_t)(B + b) * D;

    float u2 = 0.f, v2 = 0.f, duv = 0.f;
    for (int d = 0; d < D; ++d) {
        float a = u[d], c = v[d];
        u2 += a * a; v2 += c * c; duv += a * c;
    }

    // angle term
    float nv     = sqrtf(v2);
    float euclid = sqrtf(fmaxf(u2 - 2.0f * duv + v2, 0.0f));
    float rad    = fmaxf(1.0f + u2 * v2 - 2.0f * duv, EPS_F);
    float den    = fmaxf(nv * euclid * sqrtf(rad), EPS_F);
    float cosang = (duv * (1.0f + v2) - v2 * (1.0f + u2)) / den;
    cosang = fminf(fmaxf(cosang, -BND_F), BND_F);
    float angle = acosf(cosang);

    // negative-sampling term
    float dsq    = pdist_from_dots(u2, v2, duv);
    float expneg = expf(-dsq);
    float Z1 = 0.f;
    for (int j = 0; j < NEG; ++j) {
        const float* un = U + (size_t)(2 * B + b * NEG + j) * D;
        float un2 = 0.f, dun = 0.f;
        for (int d = 0; d < D; ++d) {
            float a = u[d], c = un[d];
            un2 += c * c; dun += a * c;
        }
        Z1 += expf(-pdist_from_dots(u2, un2, dun));
    }
    float ns = -logf(expneg / (Z1 + expneg));

    __shared__ float red[64];
    red[b] = angle + ns;   // alpha = 0.5 => 2(1-a)*angle + 2a*ns = angle + ns
    __syncthreads();
    for (int off = 32; off > 0; off >>= 1) {
        if (b < off) red[b] += red[b + off];
        __syncthreads();
    }
    if (b == 0) out[0] = red[0] / (float)B;
}

// ---------------------------------------------------------------------------
extern "C" void kernel_launch(void* const* d_in, const int* in_sizes, int n_in,
                              void* d_out, int out_size, void* d_ws, size_t ws_size,
                              hipStream_t stream) {
    const float* encoded   = (const float*)d_in[0];  // [64,256,768]
    const float* n_encoded = (const float*)d_in[1];  // [256,256,768]
    const float* mask1     = (const float*)d_in[2];  // [64,256,1]
    const float* mask2     = (const float*)d_in[3];  // [64,256,1]
    const float* maskn     = (const float*)d_in[4];  // [256,256,1]
    const float* W         = (const float*)d_in[5];  // [64,768]

    const int B = 64, S = 256, H = 768, D = 64, NEG = 4;
    const int R = 2 * B + B * NEG;   // 384 selected rows

    float* Y  = (float*)d_ws;              // [R, H]  = 294912 floats
    float* xn = Y + (size_t)R * H;         // [R]
    float* MX = xn + R;                    // [R, D]  = 24576 floats

    gather_expmap_kernel<<<R, S, 0, stream>>>(encoded, n_encoded, mask1, mask2,
                                              maskn, Y, xn, B, S, H);
    wmma_gemm_kernel<<<(R / 16) * (D / 16), 32, 0, stream>>>(Y, W, MX, H, D);
    mobius_scale_kernel<<<R, D, 0, stream>>>(MX, xn, D);
    final_loss_kernel<<<1, B, 0, stream>>>(MX, (float*)d_out, B, D, NEG);
}